// Ranker_63161789055239
// MI455X (gfx1250) — compile-verified
//
#include <hip/hip_runtime.h>
#include <hip/hip_bf16.h>

typedef __attribute__((ext_vector_type(2))) float v2f;
typedef __attribute__((ext_vector_type(8))) float v8f;

#define ROW_PITCH 1028
#define AJ_OFS    (16 * ROW_PITCH)
#define W3_OFS    (32 * ROW_PITCH)
#define W4_OFS    (W3_OFS + 3 * 1024)
#define TILE_OFS  (W4_OFS + 3 * 1024)
#define LDS_FLOATS (TILE_OFS + 3 * 256)
#define LDS_BYTES  (LDS_FLOATS * 4)   // 159232 B < 320 KB/WGP

__global__ void rank_rows(const float* __restrict__ Mp,
                          const float* __restrict__ W,
                          float* __restrict__ R) {
    const int b = blockIdx.x;
    const int t = threadIdx.x;             // 0..383
    const int kind = (t >= 192) ? 1 : 0;   // 0 -> R2 (W2+W3), 1 -> R1 (W1+W3)
    const int u = t - kind * 192;
    const int r = u / 3;
    const int c = u - r * 3;

    const float* row = Mp + ((size_t)b * 64 + r) * 1024;
    const float* w12 = W + c * 4096 + (kind ? 0 : 1024);
    const float* w3  = W + c * 4096 + 2048;

    float acc = 0.f;
    for (int d = 0; d < 1024; d += 4) {
        float4 m  = *(const float4*)(row + d);
        float4 wa = *(const float4*)(w12 + d);
        float4 wb = *(const float4*)(w3 + d);
        acc = fmaf(m.x, wa.x + wb.x, acc);
        acc = fmaf(m.y, wa.y + wb.y, acc);
        acc = fmaf(m.z, wa.z + wb.z, acc);
        acc = fmaf(m.w, wa.w + wb.w, acc);
    }
    R[(size_t)kind * 3072 + ((size_t)b * 64 + r) * 3 + c] = acc;
}

__global__ void pair_tile(const float* __restrict__ Mp,
                          const float* __restrict__ W,
                          const float* __restrict__ R,
                          float* __restrict__ out) {
    extern __shared__ float lds[];

    const int tid = threadIdx.x;
    const int jt = blockIdx.x, it = blockIdx.y, b = blockIdx.z;
    const int i = tid >> 4;
    const int j = tid & 15;

    // ---- Phase 0: stage 32 rows + W3/W4 into LDS, zero reduction tile ----
    const float* MpB = Mp + (size_t)b * 64 * 1024;
    #pragma unroll
    for (int v = 0; v < 32; ++v) {                   // constant trip count
        const int idx = tid + v * 256;               // 0..8191
        const int row = idx >> 6;
        const int c4  = idx & 63;
        const int grow = (row < 16) ? (it * 16 + row) : (jt * 16 + row - 16);
        float4 val = *(const float4*)(MpB + (size_t)grow * 1024 + c4 * 4);
        *(float4*)(lds + row * ROW_PITCH + c4 * 4) = val;
    }
    #pragma unroll
    for (int v = 0; v < 6; ++v) {                    // W3 then W4, 1536 float4
        const int idx = tid + v * 256;
        const int which = idx / 768;                 // 0: W3, 1: W4
        const int rr = idx - which * 768;
        const int c = rr >> 8;
        const int d4 = rr & 255;
        float4 val = *(const float4*)(W + c * 4096 + 2048 + which * 1024 + d4 * 4);
        *(float4*)(lds + (which ? W4_OFS : W3_OFS) + c * 1024 + d4 * 4) = val;
    }
    #pragma unroll
    for (int v = 0; v < 3; ++v)
        lds[TILE_OFS + tid + v * 256] = 0.f;
    __syncthreads();

    // ---- Phase 1: t3 core: sum_d min(a_i, a_j) * W3[c,d] ----
    float acc0 = 0.f, acc1 = 0.f, acc2 = 0.f;
    {
        const float* Ai  = lds + i * ROW_PITCH;
        const float* Aj  = lds + AJ_OFS + j * ROW_PITCH;
        const float* W3l = lds + W3_OFS;
        for (int d = 0; d < 1024; d += 4) {
            float4 a  = *(const float4*)(Ai + d);
            float4 bq = *(const float4*)(Aj + d);
            float m0 = fminf(a.x, bq.x), m1 = fminf(a.y, bq.y);
            float m2 = fminf(a.z, bq.z), m3 = fminf(a.w, bq.w);
            float4 w0 = *(const float4*)(W3l + d);
            float4 w1 = *(const float4*)(W3l + 1024 + d);
            float4 w2 = *(const float4*)(W3l + 2048 + d);
            acc0 = fmaf(m3, w0.w, fmaf(m2, w0.z, fmaf(m1, w0.y, fmaf(m0, w0.x, acc0))));
            acc1 = fmaf(m3, w1.w, fmaf(m2, w1.z, fmaf(m1, w1.y, fmaf(m0, w1.x, acc1))));
            acc2 = fmaf(m3, w2.w, fmaf(m2, w2.z, fmaf(m1, w2.y, fmaf(m0, w2.x, acc2))));
        }
    }

    // ---- Phase 2: t4 = A * diag(W4_c) * A^T via V_WMMA_F32_16X16X4_F32 ----
    // Each wave handles K slice [wv*128, wv*128+128); the divergent slice
    // offset is folded into the base pointers so the loop trip count (32)
    // is compile-time constant -> no exec-masked loop around WMMA.
    // The A/B row fragments are shared across all 3 channels per k-step.
    {
        const int lane = tid & 31;
        const int wv = tid >> 5;                 // wave 0..7 -> K slice
        const int mn = lane & 15;
        const int kk = (lane >> 4) << 1;         // 0 or 2 per ISA 16x4 f32 layout
        const int base = wv * 128 + kk;
        const float* ArowK = lds + mn * ROW_PITCH + base;
        const float* BrowK = lds + AJ_OFS + mn * ROW_PITCH + base;
        const float* W4K   = lds + W4_OFS + base;

        v8f accA = {0.f, 0.f, 0.f, 0.f, 0.f, 0.f, 0.f, 0.f};
        v8f accB = {0.f, 0.f, 0.f, 0.f, 0.f, 0.f, 0.f, 0.f};
        v8f accC = {0.f, 0.f, 0.f, 0.f, 0.f, 0.f, 0.f, 0.f};

        #pragma unroll 4
        for (int kq = 0; kq < 128; kq += 4) {
            float2 av = *(const float2*)(ArowK + kq);
            float2 bv = *(const float2*)(BrowK + kq);
            float2 w0 = *(const float2*)(W4K + kq);
            float2 w1 = *(const float2*)(W4K + 1024 + kq);
            float2 w2 = *(const float2*)(W4K + 2048 + kq);
            v2f a;  a.x = av.x;           a.y = av.y;
            v2f b0; b0.x = bv.x * w0.x;   b0.y = bv.y * w0.y;
            v2f b1; b1.x = bv.x * w1.x;   b1.y = bv.y * w1.y;
            v2f b2; b2.x = bv.x * w2.x;   b2.y = bv.y * w2.y;
            accA = __builtin_amdgcn_wmma_f32_16x16x4_f32(
                false, a, false, b0, (short)0, accA, false, false);
            accB = __builtin_amdgcn_wmma_f32_16x16x4_f32(
                false, a, false, b1, (short)0, accB, false, false);
            accC = __builtin_amdgcn_wmma_f32_16x16x4_f32(
                false, a, false, b2, (short)0, accC, false, false);
        }

        // Cross-wave reduction into the shared tile (ds_add_f32).
        // C/D layout: VGPR v holds M = v + 8*(lane>=16), N = lane&15.
        const int Mbase = (lane >> 4) << 3;
        float* t0 = lds + TILE_OFS + 0 * 256 + Mbase * 16 + mn;
        float* t1 = lds + TILE_OFS + 1 * 256 + Mbase * 16 + mn;
        float* t2 = lds + TILE_OFS + 2 * 256 + Mbase * 16 + mn;
        #pragma unroll
        for (int v = 0; v < 8; ++v) {
            atomicAdd(t0 + v * 16, accA[v]);
            atomicAdd(t1 + v * 16, accB[v]);
            atomicAdd(t2 + v * 16, accC[v]);
        }
    }
    __syncthreads();

    // ---- Epilogue: combine all four terms ----
    {
        const int I = it * 16 + i;
        const int J = jt * 16 + j;
        const float* R2p = R + ((size_t)b * 64 + I) * 3;
        const float* R1p = R + 3072 + ((size_t)b * 64 + J) * 3;
        float* op = out + (((size_t)b * 64 + I) * 64 + J) * 3;
        op[0] = R1p[0] + R2p[0] - 2.f * acc0 + lds[TILE_OFS + 0 * 256 + i * 16 + j];
        op[1] = R1p[1] + R2p[1] - 2.f * acc1 + lds[TILE_OFS + 1 * 256 + i * 16 + j];
        op[2] = R1p[2] + R2p[2] - 2.f * acc2 + lds[TILE_OFS + 2 * 256 + i * 16 + j];
    }
}

extern "C" void kernel_launch(void* const* d_in, const int* in_sizes, int n_in,
                              void* d_out, int out_size, void* d_ws, size_t ws_size,
                              hipStream_t stream) {
    const float* Mp = (const float*)d_in[0];   // (16,64,1024) f32
    // d_in[1] = mask_p: all-ones boolean, mathematically unused by the reference
    const float* W  = (const float*)d_in[2];   // (3,4096) f32
    float* out = (float*)d_out;                // (16,64,64,3) f32
    float* ws  = (float*)d_ws;                 // 24 KB used: R2 then R1

    rank_rows<<<dim3(16), dim3(384), 0, stream>>>(Mp, W, ws);
    pair_tile<<<dim3(4, 4, 16), dim3(256), LDS_BYTES, stream>>>(Mp, W, ws, out);
}